// GDCN_81784767251043
// MI455X (gfx1250) — compile-verified
//
#include <hip/hip_runtime.h>
#include <hip/hip_bf16.h>

#define T_DIM 8192
#define N_DIM 4096
#define NLAYER 4
#define T_TILE 128
#define N_TILE 32
#define T_CHUNK 16
#define HALO NLAYER
#define A_LEN (T_CHUNK + HALO)  /* 20 */
#define XS_PITCH 132            /* T_TILE + HALO; 33 float4 chunks, 16B aligned rows */

typedef int v4i_t __attribute__((ext_vector_type(4)));
typedef v4i_t __attribute__((address_space(1))) *gv4_p;  // global int4*
typedef v4i_t __attribute__((address_space(3))) *lv4_p;  // LDS int4*

__device__ __forceinline__ float fast_tanh(float v) {
#if __has_builtin(__builtin_amdgcn_tanh_f32)
  return __builtin_amdgcn_tanh_f32(v);
#elif __has_builtin(__builtin_amdgcn_tanhf)
  return __builtin_amdgcn_tanhf(v);
#else
  float r;
  asm("v_tanh_f32 %0, %1" : "=v"(r) : "v"(v));
  return r;
#endif
}

// sigmoid(x) = 0.5 + 0.5*tanh(x/2): one transcendental per gate
__device__ __forceinline__ float fast_sigmoid(float v) {
  return fmaf(0.5f, fast_tanh(0.5f * v), 0.5f);
}

// 16B global -> LDS copy; CDNA5 async-to-LDS path when available
__device__ __forceinline__ void stage16(const float* g, float* l) {
#if __has_builtin(__builtin_amdgcn_global_load_async_to_lds_b128)
  __builtin_amdgcn_global_load_async_to_lds_b128((gv4_p)(void*)g, (lv4_p)(void*)l, 0, 0);
#else
  *(float4*)l = *(const float4*)g;
#endif
}

__device__ __forceinline__ void wait_stage() {
#if __has_builtin(__builtin_amdgcn_s_wait_asynccnt)
  __builtin_amdgcn_s_wait_asynccnt(0);
#else
  asm volatile("s_wait_asynccnt 0" ::: "memory");
#endif
}

// Rolling 4-layer gated chain over A_LEN register values.
// EDGE=true masks positions t>=T to zero at every layer (per-layer right pad).
template <bool EDGE>
__device__ __forceinline__ void run_chain(float (&a)[A_LEN], int lim, float s_w, float s_b,
                                          const float (&f0)[NLAYER], const float (&f1)[NLAYER],
                                          const float (&fbv)[NLAYER],
                                          const float (&g0)[NLAYER], const float (&g1)[NLAYER],
                                          const float (&gbv)[NLAYER]) {
#pragma unroll
  for (int j = 0; j < A_LEN; ++j) {
    float z0 = fmaf(s_w, a[j], s_b);
    a[j] = (!EDGE || (j < lim)) ? z0 : 0.0f;
  }
#pragma unroll
  for (int l = 0; l < NLAYER; ++l) {
#pragma unroll
    for (int j = 0; j < A_LEN - 1 - l; ++j) {
      float fp = fmaf(f1[l], a[j + 1], fmaf(f0[l], a[j], fbv[l]));
      float gp = fmaf(g1[l], a[j + 1], fmaf(g0[l], a[j], gbv[l]));
      float v = fast_tanh(fp) * fast_sigmoid(gp);
      a[j] = (!EDGE || (j < lim)) ? v : 0.0f;
    }
  }
}

__global__ __launch_bounds__(256) void wavenet_fused(
    const float* __restrict__ x, const float* __restrict__ sw, const float* __restrict__ sb,
    const float* __restrict__ fw, const float* __restrict__ fb,
    const float* __restrict__ gw, const float* __restrict__ gb, float* __restrict__ out) {
  __shared__ float xs[N_TILE * XS_PITCH];

  const int tid = threadIdx.x;
  const int t0 = blockIdx.x * T_TILE;
  const int n0 = blockIdx.y * N_TILE;

  // Stage 32 x 132 input tile: 33 float4 chunks per row, lanes walk T (coalesced),
  // issued as async global->LDS copies tracked by ASYNCcnt.
  for (int idx = tid; idx < N_TILE * 33; idx += 256) {
    int row = idx / 33;
    int chk = idx - row * 33;
    int tcol = t0 + chk * 4;
    if (tcol > T_DIM - 4) tcol = T_DIM - 4;  // stay in-buffer; garbage masked in compute
    stage16(x + (size_t)(n0 + row) * T_DIM + tcol, xs + row * XS_PITCH + chk * 4);
  }
  wait_stage();
  __syncthreads();

  // Tiny uniform weights (4 layers x 2 taps + biases)
  float f0[NLAYER], f1[NLAYER], fbv[NLAYER], g0[NLAYER], g1[NLAYER], gbv[NLAYER];
#pragma unroll
  for (int l = 0; l < NLAYER; ++l) {
    f0[l] = fw[2 * l]; f1[l] = fw[2 * l + 1]; fbv[l] = fb[l];
    g0[l] = gw[2 * l]; g1[l] = gw[2 * l + 1]; gbv[l] = gb[l];
  }
  const float s_w = sw[0], s_b = sb[0];

  const int nl = tid & (N_TILE - 1);  // row within tile (lane id)
  const int cg = tid >> 5;            // t-chunk group (wave id)
  const int tc = t0 + cg * T_CHUNK;
  const int lim = T_DIM - tc;         // j < lim  <=>  t < T

  // Pull this thread's 20-element strip (16 outputs + 4 halo) from LDS.
  float a[A_LEN];
#pragma unroll
  for (int j = 0; j < A_LEN; j += 4) {
    float4 v = *(const float4*)(xs + nl * XS_PITCH + cg * T_CHUNK + j);
    a[j + 0] = v.x; a[j + 1] = v.y; a[j + 2] = v.z; a[j + 3] = v.w;
  }

  if (lim >= A_LEN) run_chain<false>(a, lim, s_w, s_b, f0, f1, fbv, g0, g1, gbv);
  else              run_chain<true>(a, lim, s_w, s_b, f0, f1, fbv, g0, g1, gbv);

  // Transposed store out[t, n]: lanes of a wave hold 32 consecutive n at equal t,
  // so each store instruction is a contiguous 128B row. NT: write-once stream.
  float* orow = out + (size_t)tc * N_DIM + (n0 + nl);
#pragma unroll
  for (int j = 0; j < T_CHUNK; ++j)
    __builtin_nontemporal_store(a[j], orow + (size_t)j * N_DIM);
}

extern "C" void kernel_launch(void* const* d_in, const int* in_sizes, int n_in,
                              void* d_out, int out_size, void* d_ws, size_t ws_size,
                              hipStream_t stream) {
  const float* x  = (const float*)d_in[0];
  const float* sw = (const float*)d_in[1];
  const float* sb = (const float*)d_in[2];
  const float* fw = (const float*)d_in[3];
  const float* fb = (const float*)d_in[4];
  const float* gw = (const float*)d_in[5];
  const float* gb = (const float*)d_in[6];
  float* out = (float*)d_out;

  dim3 grid(T_DIM / T_TILE, N_DIM / N_TILE);  // 64 x 128 blocks
  wavenet_fused<<<grid, dim3(256), 0, stream>>>(x, sw, sb, fw, fb, gw, gb, out);
}